// HyenaBlock_82970178224751
// MI455X (gfx1250) — compile-verified
//
#include <hip/hip_runtime.h>
#include <hip/hip_bf16.h>
#include <math.h>

typedef __attribute__((ext_vector_type(2))) float v2f;
typedef __attribute__((ext_vector_type(8))) float v8f;

#define BB 4
#define SS 4096
#define DD 256
#define GG 3
#define NO 2
#define MM (BB*SS)          // 16384
#define N2 8192
#define LOGN 13

// GEMM macro-tile: 8 waves/WG, 128x64 per WG, 16x64 per wave (4 accumulators)
#define BKM 128
#define BKN 64
#define BKK 32

__device__ __forceinline__ v8f wmma4(v2f a, v2f b, v8f c) {
  // D = A(16x4) * B(4x16) + C, fp32 matrix pipe.
  return __builtin_amdgcn_wmma_f32_16x16x4_f32(false, a, false, b, (short)0, c, false, false);
}

// ---------------------------------------------------------------------------
// fp32 WMMA GEMM: C[M,N] = A[M,K] @ B[K,N] (+bias). Row-major, lda=K, ldc=N.
// grid = (M/128, N/64). Wave w owns rows [128*bx + 16w, +16), cols [64*by, +64).
// B chunk (32x64) staged in LDS and shared by the 8 waves; A fragment reused
// across 4 WMMAs in registers; next B chunk prefetched (global_prefetch_b8).
// ---------------------------------------------------------------------------
__global__ __launch_bounds__(256) void gemm_wmma_f32(
    const float* __restrict__ A, const float* __restrict__ Bm,
    const float* __restrict__ bias, float* __restrict__ C,
    int M, int N, int Kd) {
  __shared__ float Bs[BKK * BKN];   // 8 KB
  int tid = threadIdx.x;
  int wave = tid >> 5, lane = tid & 31;
  int m0 = blockIdx.x * BKM + wave * 16;
  int n0 = blockIdx.y * BKN;
  int mr = m0 + (lane & 15);
  int kb = (lane >> 4) << 1;        // 0 or 2
  int nl = lane & 15;
  v8f acc0 = {}, acc1 = {}, acc2 = {}, acc3 = {};
  for (int kc = 0; kc < Kd; kc += BKK) {
    __syncthreads();
    #pragma unroll
    for (int i = 0; i < (BKK * BKN) / 256; ++i) {
      int idx = tid + i * 256;
      int kk = idx >> 6, nn = idx & 63;
      Bs[idx] = Bm[(kc + kk) * N + n0 + nn];
    }
    if (kc + BKK < Kd)
      __builtin_prefetch(&Bm[(kc + BKK + (tid >> 6)) * N + n0 + (tid & 63)], 0, 0);
    __syncthreads();
    #pragma unroll
    for (int ks = 0; ks < BKK; ks += 4) {
      v2f a;
      a.x = A[mr * Kd + kc + ks + kb];
      a.y = A[mr * Kd + kc + ks + kb + 1];
      int r0 = (ks + kb) * BKN, r1 = (ks + kb + 1) * BKN;
      v2f b0, b1, b2, b3;
      b0.x = Bs[r0 + nl];      b0.y = Bs[r1 + nl];
      b1.x = Bs[r0 + 16 + nl]; b1.y = Bs[r1 + 16 + nl];
      b2.x = Bs[r0 + 32 + nl]; b2.y = Bs[r1 + 32 + nl];
      b3.x = Bs[r0 + 48 + nl]; b3.y = Bs[r1 + 48 + nl];
      acc0 = wmma4(a, b0, acc0);
      acc1 = wmma4(a, b1, acc1);
      acc2 = wmma4(a, b2, acc2);
      acc3 = wmma4(a, b3, acc3);
    }
  }
  int rowBase = m0 + ((lane >> 4) << 3);
  #pragma unroll
  for (int t = 0; t < 4; ++t) {
    v8f acc = (t == 0) ? acc0 : (t == 1) ? acc1 : (t == 2) ? acc2 : acc3;
    int col = n0 + t * 16 + nl;
    float bv = bias ? bias[col] : 0.f;
    #pragma unroll
    for (int r = 0; r < 8; ++r)
      C[(rowBase + r) * N + col] = acc[r] + bv;
  }
}

// ---------------------------------------------------------------------------
// conv_kernel (G,O,Ci,K) -> KB[g][c*K+k][o]  (im2col B-matrix)
// ---------------------------------------------------------------------------
__global__ void transpose_kernel_w(const float* __restrict__ ck, float* __restrict__ KB) {
  int idx = blockIdx.x * blockDim.x + threadIdx.x;
  const int total = GG * DD * DD * 3;
  if (idx >= total) return;
  int k = idx % 3; int t = idx / 3;
  int c = t % DD;  t /= DD;
  int o = t % DD;  int g = t / DD;
  KB[(g * (DD * 3) + (c * 3 + k)) * DD + o] = ck[idx];
}

// ---------------------------------------------------------------------------
// Masked grouped conv as WMMA GEMM, same macro-tiling as gemm_wmma_f32.
// A is im2col-gathered on the fly with the positions mask.
// grid = (M/128, G, N/64)
// ---------------------------------------------------------------------------
__global__ __launch_bounds__(256) void conv_gemm_wmma(
    const float* __restrict__ XP, const float* __restrict__ KB,
    const int* __restrict__ pos, float* __restrict__ Z) {
  __shared__ float Bs[BKK * BKN];
  const int Kd = DD * 3, N = DD;
  int tid = threadIdx.x;
  int wave = tid >> 5, lane = tid & 31;
  int g = blockIdx.y;
  int m0 = blockIdx.x * BKM + wave * 16;
  int n0 = blockIdx.z * BKN;
  int m = m0 + (lane & 15);
  int b = m / SS, s = m - b * SS;
  int pc = pos[b * SS + s];
  int base[3]; bool msk[3];
  #pragma unroll
  for (int k = 0; k < 3; ++k) {
    int sp = s + k - 1;
    bool ok = (sp >= 0) && (sp < SS);
    int pv = ok ? pos[b * SS + sp] : 0;
    msk[k]  = ok && (pv == pc + (k - 1));
    base[k] = ok ? ((b * SS + sp) * (GG * DD) + g * DD) : 0;
  }
  int kb = (lane >> 4) << 1;
  int nl = lane & 31 & 15;
  v8f acc0 = {}, acc1 = {}, acc2 = {}, acc3 = {};
  for (int kc = 0; kc < Kd; kc += BKK) {
    __syncthreads();
    #pragma unroll
    for (int i = 0; i < (BKK * BKN) / 256; ++i) {
      int idx = tid + i * 256;
      int kk = idx >> 6, nn = idx & 63;
      Bs[idx] = KB[(g * Kd + kc + kk) * N + n0 + nn];
    }
    if (kc + BKK < Kd)
      __builtin_prefetch(&KB[(g * Kd + kc + BKK + (tid >> 6)) * N + n0 + (tid & 63)], 0, 0);
    __syncthreads();
    #pragma unroll
    for (int ks = 0; ks < BKK; ks += 4) {
      int kk0 = kc + ks + kb, kk1 = kk0 + 1;
      int c0 = kk0 / 3, q0 = kk0 - c0 * 3;
      int c1 = kk1 / 3, q1 = kk1 - c1 * 3;
      v2f a;
      a.x = msk[q0] ? XP[base[q0] + c0] : 0.f;
      a.y = msk[q1] ? XP[base[q1] + c1] : 0.f;
      int r0 = (ks + kb) * BKN, r1 = (ks + kb + 1) * BKN;
      v2f b0, b1, b2, b3;
      b0.x = Bs[r0 + nl];      b0.y = Bs[r1 + nl];
      b1.x = Bs[r0 + 16 + nl]; b1.y = Bs[r1 + 16 + nl];
      b2.x = Bs[r0 + 32 + nl]; b2.y = Bs[r1 + 32 + nl];
      b3.x = Bs[r0 + 48 + nl]; b3.y = Bs[r1 + 48 + nl];
      acc0 = wmma4(a, b0, acc0);
      acc1 = wmma4(a, b1, acc1);
      acc2 = wmma4(a, b2, acc2);
      acc3 = wmma4(a, b3, acc3);
    }
  }
  int rowBase = m0 + ((lane >> 4) << 3);
  #pragma unroll
  for (int t = 0; t < 4; ++t) {
    v8f acc = (t == 0) ? acc0 : (t == 1) ? acc1 : (t == 2) ? acc2 : acc3;
    int col = n0 + t * 16 + nl;
    #pragma unroll
    for (int r = 0; r < 8; ++r)
      Z[(rowBase + r) * (GG * DD) + g * DD + col] = acc[r];
  }
}

// ---------------------------------------------------------------------------
// Reference's raw reshape (B,S,G,D)->(B,D,G,S): streams[g][b,d,s] = flat remap.
// ---------------------------------------------------------------------------
__global__ void scramble_kernel(const float* __restrict__ Z, float* __restrict__ S0,
                                float* __restrict__ S1, float* __restrict__ S2) {
  int idx = blockIdx.x * blockDim.x + threadIdx.x;
  const int total = BB * DD * GG * SS;
  if (idx >= total) return;
  int s = idx % SS; int t = idx / SS;
  int g = t % GG;   t /= GG;
  int d = t % DD;   int b = t / DD;
  int f   = (d * GG + g) * SS + s;       // flat within (D,G,S) view
  int sp  = f / (GG * DD);
  int rem = f - sp * (GG * DD);
  float v = Z[(b * SS + sp) * (GG * DD) + rem];
  float* dst = (g == 0) ? S0 : (g == 1) ? S1 : S2;
  dst[(b * DD + d) * SS + s] = v;
}

// ---------------------------------------------------------------------------
// RoPE
// ---------------------------------------------------------------------------
__global__ void rope_kernel(const float* __restrict__ emb, const int* __restrict__ pos,
                            float* __restrict__ T) {
  int idx = blockIdx.x * blockDim.x + threadIdx.x;
  const int total = MM * (DD / 2);
  if (idx >= total) return;
  int r = idx / (DD / 2), d2 = idx % (DD / 2);
  float p = (float)pos[r];
  float theta = __expf(-((float)(2 * d2) / (float)DD) * 9.210340371976184f); // ln(1e4)
  float sn, cs; __sincosf(p * theta, &sn, &cs);
  float e0 = emb[r * DD + 2 * d2], e1 = emb[r * DD + 2 * d2 + 1];
  T[r * DD + 2 * d2]     = e0 * cs - e1 * sn;
  T[r * DD + 2 * d2 + 1] = e0 * sn + e1 * cs;
}

// ---------------------------------------------------------------------------
// LayerNorm + exact GELU, in-place, one row (D=256) per 256-thread block.
// ---------------------------------------------------------------------------
__global__ __launch_bounds__(256) void ln_gelu_kernel(
    float* __restrict__ H, const float* __restrict__ g, const float* __restrict__ b) {
  __shared__ float red[256];
  int r = blockIdx.x, d = threadIdx.x;
  float x = H[r * DD + d];
  red[d] = x; __syncthreads();
  for (int off = 128; off; off >>= 1) { if (d < off) red[d] += red[d + off]; __syncthreads(); }
  float mu = red[0] * (1.0f / DD);
  __syncthreads();
  float xm = x - mu;
  red[d] = xm * xm; __syncthreads();
  for (int off = 128; off; off >>= 1) { if (d < off) red[d] += red[d + off]; __syncthreads(); }
  float var = red[0] * (1.0f / DD);
  float y = xm * rsqrtf(var + 1e-5f) * g[d] + b[d];
  H[r * DD + d] = 0.5f * y * (1.0f + erff(y * 0.70710678118654752f));
}

// ---------------------------------------------------------------------------
// L1-normalize over D and transpose to HT[((b*NO+i)*D+d)*S + s]
// ---------------------------------------------------------------------------
__global__ __launch_bounds__(256) void l1norm_t_kernel(
    const float* __restrict__ HH, float* __restrict__ HT) {
  __shared__ float red[256];
  int bid = blockIdx.x;
  int i = bid % NO; int t = bid / NO;
  int s = t % SS;   int b = t / SS;
  int d = threadIdx.x;
  float v = HH[(b * SS + s) * (NO * DD) + i * DD + d];
  red[d] = fabsf(v); __syncthreads();
  for (int off = 128; off; off >>= 1) { if (d < off) red[d] += red[d + off]; __syncthreads(); }
  float l1 = red[0] + 1e-8f;
  HT[((b * NO + i) * DD + d) * SS + s] = v / l1;
}

// ---------------------------------------------------------------------------
// In-place radix-2 DIT FFT over 8192 complex values in LDS (forward, e^{-i..})
// ---------------------------------------------------------------------------
__device__ void fft8192(float2* zb, int tid) {
  for (int i = tid; i < N2; i += 256) {
    int j = (int)(__brev((unsigned)i) >> (32 - LOGN));
    if (j > i) { float2 t = zb[i]; zb[i] = zb[j]; zb[j] = t; }
  }
  __syncthreads();
  for (int st = 1; st <= LOGN; ++st) {
    int mh = 1 << (st - 1);
    for (int t = tid; t < N2 / 2; t += 256) {
      int grp = t >> (st - 1);
      int pos = t & (mh - 1);
      int i0 = (grp << st) + pos;
      int i1 = i0 + mh;
      float ang = -3.14159265358979323846f * (float)pos / (float)mh;
      float sn, cs; __sincosf(ang, &sn, &cs);
      float2 a = zb[i0], b = zb[i1];
      float tr = b.x * cs - b.y * sn;
      float ti = b.x * sn + b.y * cs;
      zb[i0] = make_float2(a.x + tr, a.y + ti);
      zb[i1] = make_float2(a.x - tr, a.y - ti);
    }
    __syncthreads();
  }
}

// ---------------------------------------------------------------------------
// One (b,d) row: FFT-long-conv(v, h_i) then v_new = xs * (v*Bp + conv/n^2).
// Packs v,h into one complex forward FFT, Hermitian extraction, conj-trick IFFT.
// ---------------------------------------------------------------------------
__global__ __launch_bounds__(256) void fftconv_kernel(
    const float* __restrict__ Vin, const float* __restrict__ HT,
    const float* __restrict__ XS, const float* __restrict__ Bp,
    float* __restrict__ Vout, int order, int transposedOut) {
  __shared__ float2 zb[N2];   // 64 KB of the 320 KB WGP LDS
  int row = blockIdx.x;       // b*D + d
  int b = row / DD, d = row % DD;
  int tid = threadIdx.x;
  const float* vrow = Vin + row * SS;
  const float* hrow = HT + ((b * NO + order) * DD + d) * SS;
  for (int s = tid; s < N2; s += 256) {
    float vr = (s < SS) ? vrow[s] : 0.f;
    float hr = (s < SS) ? hrow[s] : 0.f;
    zb[s] = make_float2(vr, hr);
  }
  __syncthreads();
  fft8192(zb, tid);
  // spectral multiply: X=FFT(v), Y=FFT(h) from packed Z; store conj(P=XY)
  for (int k = tid; k <= N2 / 2; k += 256) {
    if (k == 0) {
      float p = zb[0].x * zb[0].y;
      zb[0] = make_float2(p, 0.f);
    } else if (k == N2 / 2) {
      float p = zb[k].x * zb[k].y;
      zb[k] = make_float2(p, 0.f);
    } else {
      float2 Zk = zb[k], Zm = zb[N2 - k];
      float Xr = 0.5f * (Zk.x + Zm.x), Xi = 0.5f * (Zk.y - Zm.y);
      float Yr = 0.5f * (Zk.y + Zm.y), Yi = 0.5f * (Zm.x - Zk.x);
      float Pr = Xr * Yr - Xi * Yi;
      float Pi = Xr * Yi + Xi * Yr;
      zb[k]      = make_float2(Pr, -Pi);   // conj(P_k)
      zb[N2 - k] = make_float2(Pr,  Pi);   // conj(P_{n-k}) = P_k
    }
  }
  __syncthreads();
  fft8192(zb, tid);          // DFT(conj P); Re == Re(IDFT_raw(P))
  const float scale = 1.0f / ((float)N2 * (float)N2); // forward-norm semantics
  float bp = Bp[order * DD + d];
  for (int s = tid; s < SS; s += 256) {
    float conv = zb[s].x * scale;
    float vnew = XS[row * SS + s] * (vrow[s] * bp + conv);
    if (transposedOut) Vout[(b * SS + s) * DD + d] = vnew;
    else               Vout[row * SS + s] = vnew;
  }
}

// ---------------------------------------------------------------------------
extern "C" void kernel_launch(void* const* d_in, const int* in_sizes, int n_in,
                              void* d_out, int out_size, void* d_ws, size_t ws_size,
                              hipStream_t stream) {
  const float* emb  = (const float*)d_in[0];
  const int*   pos  = (const int*)d_in[1];
  const float* Wprj = (const float*)d_in[2];
  const float* ck   = (const float*)d_in[3];
  const float* W1   = (const float*)d_in[4];
  const float* b1   = (const float*)d_in[5];
  const float* gam  = (const float*)d_in[6];
  const float* bet  = (const float*)d_in[7];
  const float* W2   = (const float*)d_in[8];
  const float* b2   = (const float*)d_in[9];
  const float* Wout = (const float*)d_in[10];
  const float* Bp   = (const float*)d_in[11];
  float* ws = (float*)d_ws;

  // workspace layout (floats), with lifetime-safe region reuse (~170 MB total)
  float* XP = ws + 0;              // 12,582,912   (dead after conv)
  float* H1 = ws + 0;              //  4,194,304   (reuses XP)
  float* HH = ws + 4194304;        //  8,388,608   (reuses XP)
  float* Z  = ws + 12582912;       // 12,582,912   (dead after scramble)
  float* HT = ws + 12582912;       //  8,388,608   (reuses Z)
  float* T  = ws + 20971520;       //  4,194,304   (reuses Z tail; dead after FFN1)
  float* V2 = ws + 20971520;       //  4,194,304   (reuses T slot)
  float* KB = ws + 25165824;       //    589,824
  float* S0 = ws + 25755648;       //  4,194,304
  float* S1 = ws + 29949952;       //  4,194,304
  float* V  = ws + 34144256;       //  4,194,304
  float* VT = ws + 38338560;       //  4,194,304
  float* OUT = (float*)d_out;

  // 1) conv weight -> im2col B-matrix
  transpose_kernel_w<<<(GG*DD*DD*3 + 255)/256, 256, 0, stream>>>(ck, KB);
  // 2) xproj = emb @ W_proj   (16384 x 768, K=256)
  {
    dim3 grid(MM/BKM, (GG*DD)/BKN);
    gemm_wmma_f32<<<grid, 256, 0, stream>>>(emb, Wprj, nullptr, XP, MM, GG*DD, DD);
  }
  // 3) masked grouped conv as WMMA GEMM (per group)
  {
    dim3 grid(MM/BKM, GG, DD/BKN);
    conv_gemm_wmma<<<grid, 256, 0, stream>>>(XP, KB, pos, Z);
  }
  // 4) reference reshape scramble -> streams S0,S1 and initial V
  scramble_kernel<<<(BB*DD*GG*SS + 255)/256, 256, 0, stream>>>(Z, S0, S1, V);
  // 5) RoPE -> T
  rope_kernel<<<(MM*(DD/2) + 255)/256, 256, 0, stream>>>(emb, pos, T);
  // 6) FFN: h = LN(T@W1+b1)*g+b, gelu
  {
    dim3 grid(MM/BKM, DD/BKN);
    gemm_wmma_f32<<<grid, 256, 0, stream>>>(T, W1, b1, H1, MM, DD, DD);
  }
  ln_gelu_kernel<<<MM, 256, 0, stream>>>(H1, gam, bet);
  // 7) HH = h @ W2 + b2   (16384 x 512)
  {
    dim3 grid(MM/BKM, (NO*DD)/BKN);
    gemm_wmma_f32<<<grid, 256, 0, stream>>>(H1, W2, b2, HH, MM, NO*DD, DD);
  }
  // 8) L1 normalize over D, transpose to (B,NO,D,S)
  l1norm_t_kernel<<<BB*SS*NO, 256, 0, stream>>>(HH, HT);
  // 9) order 0: V2 = S0 * (V*Bp0 + fftconv(V, h0))
  fftconv_kernel<<<BB*DD, 256, 0, stream>>>(V, HT, S0, Bp, V2, 0, 0);
  // 10) order 1: VT[(b,s,d)] = S1 * (V2*Bp1 + fftconv(V2, h1))
  fftconv_kernel<<<BB*DD, 256, 0, stream>>>(V2, HT, S1, Bp, VT, 1, 1);
  // 11) out = VT @ out_proj
  {
    dim3 grid(MM/BKM, DD/BKN);
    gemm_wmma_f32<<<grid, 256, 0, stream>>>(VT, Wout, nullptr, OUT, MM, DD, DD);
  }
}